// TraditionalTripletLoss_82746839925071
// MI455X (gfx1250) — compile-verified
//
#include <hip/hip_runtime.h>
#include <math.h>

// Problem constants (match the JAX reference)
#define BN 8192
#define DK 128
#define MARGIN 0.1f
#define F32_INF_BITS 0x7f800000u

typedef float v2f __attribute__((ext_vector_type(2)));
typedef float v8f __attribute__((ext_vector_type(8)));

// ---------------------------------------------------------------------------
// Kernel 1: squared row norms + init of hardest-pos / hardest-neg accumulators
// 32 lanes per row (wave32), 8 rows per 256-thread block.
// ---------------------------------------------------------------------------
__global__ __launch_bounds__(256) void sq_init_kernel(
    const float* __restrict__ x, float* __restrict__ sq,
    unsigned* __restrict__ hp, unsigned* __restrict__ hn) {
  const int lane = threadIdx.x & 31;
  const int row = blockIdx.x * 8 + (threadIdx.x >> 5);
  const float4 v = ((const float4*)(x + row * DK))[lane];  // 32 lanes * 4 = 128
  float s = v.x * v.x + v.y * v.y + v.z * v.z + v.w * v.w;
#pragma unroll
  for (int m = 16; m >= 1; m >>= 1) s += __shfl_xor(s, m, 32);
  if (lane == 0) {
    sq[row] = s;
    hp[row] = 0u;            // max over positives (all distances >= 0)
    hn[row] = F32_INF_BITS;  // min over negatives
  }
}

// ---------------------------------------------------------------------------
// Kernel 2: fused  G = X X^T  (f32 WMMA) -> distances -> batch-hard mining.
// Symmetry: only tiles with colBase+31 >= rowBase are launched; each tile
// mines BOTH its rows (anchors = rows) and its columns (anchors = cols).
// Duplicate contributions near the diagonal are harmless for max/min atomics.
// Block = 128 threads = 4 waves; block tile = 64 rows x 32 cols of G.
// Each wave owns a 16x32 strip: two 16x16 WMMA tiles sharing the A fragment
// (3 LDS b64 reads feed 2 wmmas -> 1.5 LDS reads per matrix op instead of 2).
// ---------------------------------------------------------------------------
__global__ __launch_bounds__(128) void triplet_tile_kernel(
    const float* __restrict__ x, const int* __restrict__ labels,
    const float* __restrict__ sq, unsigned* __restrict__ hp,
    unsigned* __restrict__ hn) {
  // Padded LDS stride: 132 floats = 528 B per row -> rows 16B-aligned and a
  // 4-bank advance per row; a fragment read (16 rows x b64) covers 64 banks.
  constexpr int LSTR = 132;
  __shared__ float As[64 * LSTR];  // 64 G-rows x K=128
  __shared__ float Bs[32 * LSTR];  // 32 G-cols x K=128

  const int rowBase = blockIdx.y * 64;
  const int colBase = blockIdx.x * 32;
  if (colBase + 32 <= rowBase) return;  // strictly-lower tile: mirror covers it

  const int tid = threadIdx.x;

  // Stage A tile: 2 threads/row, 16 float4 each (global_load_b128 -> ds_store_b128)
  {
    const int r = tid >> 1, q = tid & 1;
    const float4* src = (const float4*)(x + (rowBase + r) * DK + q * 64);
    float4* dst = (float4*)(As + r * LSTR + q * 64);
#pragma unroll
    for (int i = 0; i < 16; ++i) dst[i] = src[i];
  }
  // Stage B tile: 4 threads/row, 8 float4 each
  {
    const int r = tid >> 2, q = tid & 3;
    const float4* src = (const float4*)(x + (colBase + r) * DK + q * 32);
    float4* dst = (float4*)(Bs + r * LSTR + q * 32);
#pragma unroll
    for (int i = 0; i < 8; ++i) dst[i] = src[i];
  }
  __syncthreads();

  const int wm = tid >> 5;  // wave id = row sub-tile 0..3
  const int lane = tid & 31;
  const int laneM = lane & 15;
  const int half = lane >> 4;

  // f32 16x4 fragment layout (ISA 7.12.2): VGPR pair steps K by 1,
  // lane halves step K by 2; B mirrored with N = lane&15.
  const float* aRow = As + (wm * 16 + laneM) * LSTR + 2 * half;
  const float* b0Row = Bs + laneM * LSTR + 2 * half;         // cols 0..15
  const float* b1Row = Bs + (16 + laneM) * LSTR + 2 * half;  // cols 16..31

  v8f c0 = {}, c1 = {};
#pragma unroll
  for (int kk = 0; kk < 32; ++kk) {  // K = 128 in steps of 4
    v2f a = *(const v2f*)(aRow + kk * 4);
    v2f b0 = *(const v2f*)(b0Row + kk * 4);
    v2f b1 = *(const v2f*)(b1Row + kk * 4);
    c0 = __builtin_amdgcn_wmma_f32_16x16x4_f32(false, a, false, b0, (short)0,
                                               c0, false, false);
    c1 = __builtin_amdgcn_wmma_f32_16x16x4_f32(false, a, false, b1, (short)0,
                                               c1, false, false);
  }

  // Epilogue. C/D layout: lane holds N = lane&15, rows M = v + 8*half.
  const int colG0 = colBase + laneM;
  const int colG1 = colBase + 16 + laneM;
  const float sqc0 = sq[colG0], sqc1 = sq[colG1];
  const int lc0 = labels[colG0], lc1 = labels[colG1];
  const float inf = __uint_as_float(F32_INF_BITS);

  float cpos0 = 0.0f, cneg0 = inf;  // column-anchor accumulators (tile 0)
  float cpos1 = 0.0f, cneg1 = inf;  // column-anchor accumulators (tile 1)

#pragma unroll
  for (int v = 0; v < 8; ++v) {
    const int rowG = rowBase + wm * 16 + v + 8 * half;
    const float sqr = sq[rowG];
    const int lr = labels[rowG];

    float d2a = sqr + sqc0 - 2.0f * c0[v];
    const float d0 = d2a > 0.0f ? sqrtf(d2a) : 0.0f;
    const bool s0 = (lr == lc0);
    const float p0 = (s0 && d0 > 0.0f) ? d0 : 0.0f;  // excludes self (d==0)
    const float n0 = s0 ? inf : d0;

    float d2b = sqr + sqc1 - 2.0f * c1[v];
    const float d1 = d2b > 0.0f ? sqrtf(d2b) : 0.0f;
    const bool s1 = (lr == lc1);
    const float p1 = (s1 && d1 > 0.0f) ? d1 : 0.0f;
    const float n1 = s1 ? inf : d1;

    // ---- row mining: both tiles share this row; reduce 32 cols ----
    float rp = fmaxf(p0, p1);
    float rn = fminf(n0, n1);
#pragma unroll
    for (int m = 1; m < 16; m <<= 1) {  // stays within the half-wave
      rp = fmaxf(rp, __shfl_xor(rp, m, 32));
      rn = fminf(rn, __shfl_xor(rn, m, 32));
    }
    if (laneM == v) {  // one lane per row publishes (spread across lanes)
      if (rp > 0.0f) atomicMax(&hp[rowG], __float_as_uint(rp));
      if (rn < inf) atomicMin(&hn[rowG], __float_as_uint(rn));
    }

    // ---- column mining: accumulate the 8 rows this lane sees ----
    cpos0 = fmaxf(cpos0, p0);
    cneg0 = fminf(cneg0, n0);
    cpos1 = fmaxf(cpos1, p1);
    cneg1 = fminf(cneg1, n1);
  }

  // Combine the two half-waves (other 8 rows of each column).
  cpos0 = fmaxf(cpos0, __shfl_xor(cpos0, 16, 32));
  cneg0 = fminf(cneg0, __shfl_xor(cneg0, 16, 32));
  cpos1 = fmaxf(cpos1, __shfl_xor(cpos1, 16, 32));
  cneg1 = fminf(cneg1, __shfl_xor(cneg1, 16, 32));
  if (half == 0) {  // lanes 0-15 publish tile-0 columns
    if (cpos0 > 0.0f) atomicMax(&hp[colG0], __float_as_uint(cpos0));
    if (cneg0 < inf) atomicMin(&hn[colG0], __float_as_uint(cneg0));
  } else {  // lanes 16-31 publish tile-1 columns
    if (cpos1 > 0.0f) atomicMax(&hp[colG1], __float_as_uint(cpos1));
    if (cneg1 < inf) atomicMin(&hn[colG1], __float_as_uint(cneg1));
  }
}

// ---------------------------------------------------------------------------
// Kernel 3: per-anchor hinge + mean over valid anchors (single block).
// ---------------------------------------------------------------------------
__global__ __launch_bounds__(256) void finalize_kernel(
    const unsigned* __restrict__ hp, const unsigned* __restrict__ hn,
    float* __restrict__ out) {
  __shared__ float ssum[256];
  __shared__ int scnt[256];
  const float inf = __uint_as_float(F32_INF_BITS);
  float sum = 0.0f;
  int cnt = 0;
  for (int i = threadIdx.x; i < BN; i += 256) {
    const float p = __uint_as_float(hp[i]);
    const float n = __uint_as_float(hn[i]);
    if (p > 0.0f && n < inf) {  // row has >=1 positive and >=1 negative
      sum += fmaxf(p - n + MARGIN, 0.0f);
      cnt += 1;
    }
  }
  ssum[threadIdx.x] = sum;
  scnt[threadIdx.x] = cnt;
  __syncthreads();
  for (int s = 128; s > 0; s >>= 1) {
    if (threadIdx.x < (unsigned)s) {
      ssum[threadIdx.x] += ssum[threadIdx.x + s];
      scnt[threadIdx.x] += scnt[threadIdx.x + s];
    }
    __syncthreads();
  }
  if (threadIdx.x == 0) out[0] = ssum[0] / fmaxf((float)scnt[0], 1.0f);
}

// ---------------------------------------------------------------------------
extern "C" void kernel_launch(void* const* d_in, const int* in_sizes, int n_in,
                              void* d_out, int out_size, void* d_ws,
                              size_t ws_size, hipStream_t stream) {
  (void)in_sizes; (void)n_in; (void)out_size; (void)ws_size;
  const float* x = (const float*)d_in[0];   // [8192,128] f32
  const int* labels = (const int*)d_in[1];  // [8192] i32
  float* out = (float*)d_out;               // scalar f32

  // Workspace: hp bits [BN] | hn bits [BN] | sq [BN]   (96 KB total)
  unsigned* hp = (unsigned*)d_ws;
  unsigned* hn = hp + BN;
  float* sqn = (float*)(hn + BN);

  sq_init_kernel<<<BN / 8, 256, 0, stream>>>(x, sqn, hp, hn);

  // Upper-triangle-intersecting tiles only (early-out inside the kernel).
  dim3 grid(BN / 32, BN / 64);  // (256, 128) tiles of 64x32
  triplet_tile_kernel<<<grid, 128, 0, stream>>>(x, labels, sqn, hp, hn);

  finalize_kernel<<<1, 256, 0, stream>>>(hp, hn, out);
}